// TorchBiRNN_78580721647902
// MI455X (gfx1250) — compile-verified
//
#include <hip/hip_runtime.h>
#include <math.h>

typedef float v2f __attribute__((ext_vector_type(2)));
typedef float v8f __attribute__((ext_vector_type(8)));
typedef int   v4i __attribute__((vector_size(16)));   // gcc-style, matches builtin

#define ISZ 128
#define HSZ 256
#define BSZ 64
#define TSZ 512
#define OSTRIDE (2 * HSZ)                       // floats per (b,t) row of out
#define OUT_ELEMS ((size_t)BSZ * TSZ * OSTRIDE) // ys portion of d_out
#define LDW 260  // padded LDS row stride: 16B-aligned rows, conflict-free wave32 reads

#if defined(__has_builtin)
#  if __has_builtin(__builtin_amdgcn_global_load_async_to_lds_b128)
#    define HAVE_ASYNC_LDS 1
#  else
#    define HAVE_ASYNC_LDS 0
#  endif
#else
#  define HAVE_ASYNC_LDS 0
#endif

typedef __attribute__((address_space(1))) v4i* gvec_p;  // global int4*
typedef __attribute__((address_space(3))) v4i* lvec_p;  // LDS int4*

// ---------------------------------------------------------------------------
// Kernel A: pre[d][t][b][n] = x[b][t][:] . w_ih[n][:] + b_ih[n] + b_hh[n]
// Stored in-place into out[b][t][d*HSZ + n]. 8 waves/block, one 16x16 tile
// per wave, f32 WMMA 16x16x4 over K=128.
// ---------------------------------------------------------------------------
__global__ __launch_bounds__(256) void birnn_proj(
    const float* __restrict__ x,
    const float* __restrict__ w_ih_f, const float* __restrict__ b_ih_f,
    const float* __restrict__ b_hh_f,
    const float* __restrict__ w_ih_b, const float* __restrict__ b_ih_b,
    const float* __restrict__ b_hh_b,
    float* __restrict__ out)
{
    const int tid  = threadIdx.x;
    const int wave = tid >> 5;
    const int lane = tid & 31;
    const int m0   = ((int)blockIdx.x * 8 + wave) * 16; // tile over m = t*BSZ + b
    const int n0   = (int)blockIdx.y * 16;              // tile over hidden dim
    const int d    = (int)blockIdx.z;                   // direction

    const float* w_ih = d ? w_ih_b : w_ih_f;
    const float* b_ih = d ? b_ih_b : b_ih_f;
    const float* b_hh = d ? b_hh_b : b_hh_f;

    const int half = lane >> 4;     // which K-pair this half-wave holds
    const int l16  = lane & 15;

    // A-frag source row (same logical M row for both half-waves)
    const int m  = m0 + l16;
    const int t  = m >> 6;          // m / BSZ
    const int b  = m & 63;          // m % BSZ
    const float* xrow = x + ((size_t)b * TSZ + t) * ISZ;

    // B-frag source row: B[k][n] = w_ih[n][k]
    const int n = n0 + l16;
    const float* wrow = w_ih + (size_t)n * ISZ;

    v8f c = {};
    #pragma unroll 4
    for (int k0 = 0; k0 < ISZ; k0 += 4) {
        const int kb = k0 + half * 2;
        v2f a, bm;
        a.x  = xrow[kb];  a.y  = xrow[kb + 1];
        bm.x = wrow[kb];  bm.y = wrow[kb + 1];
        c = __builtin_amdgcn_wmma_f32_16x16x4_f32(
                false, a, false, bm, (short)0, c, false, false);
    }

    const float bias = b_ih[n] + b_hh[n];

    // C/D layout: VGPR r, lanes 0-15 -> M=r ; lanes 16-31 -> M=r+8 ; N = l16
    #pragma unroll
    for (int r = 0; r < 8; ++r) {
        const int mrow = m0 + r + half * 8;
        const int tt   = mrow >> 6;
        const int bb   = mrow & 63;
        out[((size_t)bb * TSZ + tt) * OSTRIDE + d * HSZ + n] = c[r] + bias;
    }
}

// ---------------------------------------------------------------------------
// Kernel B: sequential scan. grid.x = 8 : (dir[0..1] x batch-tile[0..3]).
// 256 threads = 8 wave32; each wave owns two 16-wide N tiles.
// W_hh staged into 320KB LDS via async-DMA (ASYNCcnt path) and held for all
// 512 steps; h lives in LDS; pre C-tiles are register double-buffered so the
// L2 read latency hides under the barrier. 128 WMMA / wave / step.
// ---------------------------------------------------------------------------
__global__ __launch_bounds__(256) void birnn_scan(
    const float* __restrict__ w_hh_f,
    const float* __restrict__ w_hh_b,
    float* out)
{
    extern __shared__ float lds[];
    float* wl = lds;                  // HSZ rows x LDW : w_hh[j][k] at wl[j*LDW+k]
    float* hl = lds + HSZ * LDW;      // 16 rows x LDW  : h[b_local][k]

    const int d  = (int)blockIdx.x >> 2;        // direction
    const int b0 = ((int)blockIdx.x & 3) * 16;  // batch tile base
    const float* w_hh = d ? w_hh_b : w_hh_f;
    const int tid = threadIdx.x;

#if HAVE_ASYNC_LDS
    // Async DMA staging: 256 rows x 64 chunks of 16B, padded row stride in LDS.
    for (int chunk = tid; chunk < HSZ * (HSZ / 4); chunk += 256) {
        const int j = chunk >> 6;          // row
        const int c = (chunk & 63) << 2;   // col in floats (16B chunks)
        gvec_p src = (gvec_p)(w_hh + (size_t)j * HSZ + c);
        lvec_p dst = (lvec_p)(wl + (size_t)j * LDW + c);
        __builtin_amdgcn_global_load_async_to_lds_b128(src, dst, 0, 0);
    }
#  if __has_builtin(__builtin_amdgcn_s_wait_asynccnt)
    __builtin_amdgcn_s_wait_asynccnt(0);
#  else
    asm volatile("s_wait_asynccnt 0x0" ::: "memory");
#  endif
#else
    for (int idx = tid; idx < HSZ * HSZ; idx += 256)
        wl[(idx >> 8) * LDW + (idx & 255)] = w_hh[idx];
#endif
    for (int idx = tid; idx < 16 * LDW; idx += 256)
        hl[idx] = 0.0f;
    __syncthreads();

    const int wave = tid >> 5;
    const int lane = tid & 31;
    const int half = lane >> 4;
    const int l16  = lane & 15;
    const int n_[2] = { (wave * 2 + 0) * 16 + l16,
                        (wave * 2 + 1) * 16 + l16 };

    float cCur[2][8], cNxt[2][8], res[2][8];

    // Preload step 0's pre tile
    {
        const int t0 = d ? (TSZ - 1) : 0;
        #pragma unroll
        for (int nt = 0; nt < 2; ++nt)
            #pragma unroll
            for (int r = 0; r < 8; ++r)
                cCur[nt][r] = out[((size_t)(b0 + r + half * 8) * TSZ + t0) * OSTRIDE
                                  + d * HSZ + n_[nt]];
    }

    for (int step = 0; step < TSZ; ++step) {
        const int t = d ? (TSZ - 1 - step) : step;

        #pragma unroll
        for (int nt = 0; nt < 2; ++nt) {
            v8f c;
            #pragma unroll
            for (int r = 0; r < 8; ++r) c[r] = cCur[nt][r];

            // D = h(16x256) @ W_hh^T tile + C ;  B[k][n] = w_hh[n][k]
            const float* arow = hl + l16 * LDW;             // A row = local batch row
            const float* wrow = wl + (size_t)n_[nt] * LDW;
            #pragma unroll 8
            for (int k0 = 0; k0 < HSZ; k0 += 4) {
                const int kb = k0 + half * 2;
                v2f a, bm;
                a.x  = arow[kb];  a.y  = arow[kb + 1];
                bm.x = wrow[kb];  bm.y = wrow[kb + 1];
                c = __builtin_amdgcn_wmma_f32_16x16x4_f32(
                        false, a, false, bm, (short)0, c, false, false);
            }
            #pragma unroll
            for (int r = 0; r < 8; ++r) res[nt][r] = tanhf(c[r]);
        }

        // Prefetch next step's pre (this wave exclusively owns its (b,n) slice,
        // and t+1 != t, so this is race-free across the barrier).
        if (step + 1 < TSZ) {
            const int tn = d ? (TSZ - 2 - step) : (step + 1);
            #pragma unroll
            for (int nt = 0; nt < 2; ++nt)
                #pragma unroll
                for (int r = 0; r < 8; ++r)
                    cNxt[nt][r] = out[((size_t)(b0 + r + half * 8) * TSZ + tn) * OSTRIDE
                                      + d * HSZ + n_[nt]];
        }

        __syncthreads();   // all reads of h for this step complete

        #pragma unroll
        for (int nt = 0; nt < 2; ++nt) {
            #pragma unroll
            for (int r = 0; r < 8; ++r) {
                const int mloc = r + half * 8;
                hl[mloc * LDW + n_[nt]] = res[nt][r];
                out[((size_t)(b0 + mloc) * TSZ + t) * OSTRIDE + d * HSZ + n_[nt]]
                    = res[nt][r];
            }
        }
        __syncthreads();

        #pragma unroll
        for (int nt = 0; nt < 2; ++nt)
            #pragma unroll
            for (int r = 0; r < 8; ++r)
                cCur[nt][r] = cNxt[nt][r];
    }

    // Final hidden state -> h_n[d][b][j] appended after the ys block
    float* h_n = out + OUT_ELEMS;
    for (int idx = tid; idx < 16 * HSZ; idx += 256) {
        const int r = idx >> 8;
        const int j = idx & 255;
        h_n[((size_t)d * BSZ + (b0 + r)) * HSZ + j] = hl[r * LDW + j];
    }
}

extern "C" void kernel_launch(void* const* d_in, const int* in_sizes, int n_in,
                              void* d_out, int out_size, void* d_ws, size_t ws_size,
                              hipStream_t stream) {
    (void)in_sizes; (void)n_in; (void)out_size; (void)d_ws; (void)ws_size;

    const float* x      = (const float*)d_in[0];
    const float* w_ih_f = (const float*)d_in[1];
    const float* w_hh_f = (const float*)d_in[2];
    const float* b_ih_f = (const float*)d_in[3];
    const float* b_hh_f = (const float*)d_in[4];
    const float* w_ih_b = (const float*)d_in[5];
    const float* w_hh_b = (const float*)d_in[6];
    const float* b_ih_b = (const float*)d_in[7];
    const float* b_hh_b = (const float*)d_in[8];
    float* out = (float*)d_out;

    // Projection GEMM: 256 blocks x 8 waves, 16 N-tiles, 2 dirs.
    dim3 gA((TSZ * BSZ) / (16 * 8), HSZ / 16, 2);
    birnn_proj<<<gA, 256, 0, stream>>>(x, w_ih_f, b_ih_f, b_hh_f,
                                       w_ih_b, b_ih_b, b_hh_b, out);

    // Sequential scan: 2 dirs x 4 batch tiles; W_hh + h resident in LDS.
    const size_t smem = (size_t)(HSZ + 16) * LDW * sizeof(float); // ~276 KB
    static bool attr_set = false;
    if (!attr_set) {   // host-side, idempotent; enables >64KB dynamic LDS
        (void)hipFuncSetAttribute((const void*)birnn_scan,
                                  hipFuncAttributeMaxDynamicSharedMemorySize,
                                  (int)smem);
        attr_set = true;
    }
    birnn_scan<<<dim3(8), 256, smem, stream>>>(w_hh_f, w_hh_b, out);
}